// Model_17617955848309
// MI455X (gfx1250) — compile-verified
//
#include <hip/hip_runtime.h>
#include <hip/hip_bf16.h>

// Problem constants (match the reference).
constexpr int V = 10000;
constexpr int B = 8;
constexpr int N = 2048;

constexpr int ROWS_PER_BLOCK = 8;          // rows of the output tile per block
constexpr int THREADS        = 256;        // 8 wave32 waves
constexpr int N4             = N / 4;      // float4 chunks per row = 512
constexpr int LINES_PER_ROW  = (V * 4 + 127) / 128;  // 128B cachelines per W row = 313

typedef float v4f __attribute__((ext_vector_type(4)));
typedef int   v4i __attribute__((ext_vector_type(4)));

// out[b,i,j] = W[words[b,i], words[b,j]]  (+ root[words[b,i]] if i==j)
__global__ __launch_bounds__(THREADS)
void gather_outer_kernel(const int* __restrict__ words,
                         const float* __restrict__ W,
                         const float* __restrict__ root,
                         float* __restrict__ out) {
    __shared__ int   s_cols[N];               // column ids for this batch (8 KB)
    __shared__ int   s_rowid[ROWS_PER_BLOCK]; // row ids for this tile
    __shared__ float s_rootv[ROWS_PER_BLOCK]; // root score per tile row

    const int tiles_per_batch = N / ROWS_PER_BLOCK;      // 256
    const int b    = blockIdx.x / tiles_per_batch;
    const int tile = blockIdx.x % tiles_per_batch;
    const int row0 = tile * ROWS_PER_BLOCK;

    const int* __restrict__ wb = words + b * N;

    // Stage column indices (shared by every row of the tile) into LDS.
    for (int j = threadIdx.x; j < N; j += THREADS) {
        s_cols[j] = wb[j];
    }
    if (threadIdx.x < ROWS_PER_BLOCK) {
        const int r = wb[row0 + threadIdx.x];
        s_rowid[threadIdx.x] = r;
        s_rootv[threadIdx.x] = root[r];
    }
    __syncthreads();

    // Prefetch every cache line of the 8 needed W rows (each row ~40KB is
    // touched almost entirely by 2048 random column gathers). This streams
    // the rows into cache so the scattered dword gathers below hit.
    // 8 rows * 313 lines = 2504 prefetches spread over 256 threads.
    {
        const int total_lines = ROWS_PER_BLOCK * LINES_PER_ROW;
        for (int t = threadIdx.x; t < total_lines; t += THREADS) {
            const int rr   = t / LINES_PER_ROW;
            const int line = t % LINES_PER_ROW;
            const float* p = W + (size_t)s_rowid[rr] * V + (size_t)line * 32;
            __builtin_prefetch(p, 0, 1);   // -> global_prefetch_b8
        }
    }

    // Gather + store. Each iteration: one float4 of output.
    // Stores from a wave are 32 consecutive float4 -> 512B coalesced, and
    // non-temporal so the 134MB output stream does not evict W from L2.
    float* __restrict__ outb = out + ((size_t)b * N + row0) * (size_t)N;

    const int total4 = ROWS_PER_BLOCK * N4;              // 4096
    for (int idx = threadIdx.x; idx < total4; idx += THREADS) {
        const int rr = idx >> 9;          // idx / 512
        const int j4 = (idx & (N4 - 1)) << 2;

        const float* __restrict__ wr = W + (size_t)s_rowid[rr] * V;

        const v4i c = *reinterpret_cast<const v4i*>(&s_cols[j4]); // ds_load_b128

        v4f v;
        v.x = wr[c.x];
        v.y = wr[c.y];
        v.z = wr[c.z];
        v.w = wr[c.w];

        // Diagonal: out[b,i,i] += root[words[b,i]] where i = row0 + rr.
        const int i = row0 + rr;
        const int d = i - j4;
        if ((unsigned)d < 4u) {
            v[d] += s_rootv[rr];
        }

        v4f* dst = reinterpret_cast<v4f*>(outb + (size_t)rr * N + j4);
        __builtin_nontemporal_store(v, dst);
    }
}

extern "C" void kernel_launch(void* const* d_in, const int* in_sizes, int n_in,
                              void* d_out, int out_size, void* d_ws, size_t ws_size,
                              hipStream_t stream) {
    const int*   words = (const int*)d_in[0];   // (B, N) int32
    const float* W     = (const float*)d_in[1]; // (V, V) fp32
    const float* root  = (const float*)d_in[2]; // (V,)  fp32
    float*       out   = (float*)d_out;         // (B, N, N) fp32

    const int blocks = B * (N / ROWS_PER_BLOCK); // 2048
    gather_outer_kernel<<<blocks, THREADS, 0, stream>>>(words, W, root, out);
}